// DBGNNLayer_44573170598126
// MI455X (gfx1250) — compile-verified
//
#include <hip/hip_runtime.h>

typedef __attribute__((ext_vector_type(2))) float v2f;
typedef __attribute__((ext_vector_type(8))) float v8f;

#define N_NODES 50000
#define DIM 128
#define HID 64
#define NEDGE 500000
#define SLOPE 0.2f

// ---------------------------------------------------------------------------
// WMMA f32 GEMM:  C[rows x ncols] = act(A[rows x K] @ W[K x ncols] + bias)
// One wave32 owns a 16-row x ncols stripe: the A fragment (16x4 f32) is
// loaded once per K-step and reused across NT = ncols/16 column tiles,
// each accumulated with V_WMMA_F32_16X16X4_F32.
// A-operand layout (16x4): lanes 0-15 -> M=lane, K={v0:k0, v1:k0+1};
//                          lanes 16-31 -> M=lane-16, K={v0:k0+2, v1:k0+3}.
// B-operand layout (4x16): lanes 0-15 -> N=lane, K rows {v0:k0, v1:k0+1};
//                          lanes 16-31 -> N=lane-16, K rows {k0+2, k0+3}.
// C/D layout: VGPR j -> M = stripe*16 + j + 8*(lane>=16), N = lane&15.
// rows must be a multiple of 16 (50000 = 3125*16 here).
// ---------------------------------------------------------------------------
template <int NT>
__global__ __launch_bounds__(128) void gemm_bias_kernel(
    const float* __restrict__ A, const float* __restrict__ W,
    const float* __restrict__ bias, float* __restrict__ C,
    int rows, int K, int relu)
{
    constexpr int ncols = NT * 16;
    const int wave = (int)((blockIdx.x * blockDim.x + threadIdx.x) >> 5);
    const int lane = (int)(threadIdx.x & 31);
    if (wave * 16 >= rows) return;            // wave-uniform: EXEC stays all-1s

    const int nloc  = lane & 15;
    const int khalf = (lane >> 4) << 1;       // 0 for lanes 0-15, 2 for 16-31

    const float* __restrict__ arow = A + (size_t)(wave * 16 + nloc) * (size_t)K;

    v8f acc[NT];
#pragma unroll
    for (int t = 0; t < NT; ++t)
        acc[t] = (v8f){0.f, 0.f, 0.f, 0.f, 0.f, 0.f, 0.f, 0.f};

    for (int k0 = 0; k0 < K; k0 += 4) {
        v2f a;
        a.x = arow[k0 + khalf];
        a.y = arow[k0 + khalf + 1];
        const float* __restrict__ w0 = W + (size_t)(k0 + khalf) * ncols + nloc;
        const float* __restrict__ w1 = w0 + ncols;
#pragma unroll
        for (int t = 0; t < NT; ++t) {
            v2f b;
            b.x = w0[t * 16];                 // base + immediate offsets -> clause
            b.y = w1[t * 16];
            acc[t] = __builtin_amdgcn_wmma_f32_16x16x4_f32(
                false, a, false, b, (short)0, acc[t], false, false);
        }
    }

    const int mbase = wave * 16 + ((lane >> 4) << 3);
#pragma unroll
    for (int t = 0; t < NT; ++t) {
        const int n = t * 16 + nloc;
        const float bv = bias[n];
#pragma unroll
        for (int j = 0; j < 8; ++j) {
            float v = acc[t][j] + bv;
            if (relu) v = fmaxf(v, 0.f);
            C[(size_t)(mbase + j) * ncols + n] = v;
        }
    }
}

// ---------------------------------------------------------------------------
// Order-preserving float <-> uint mapping for atomicMax-based segment max.
// ---------------------------------------------------------------------------
__device__ __forceinline__ unsigned enc_f32(float x) {
    unsigned b = __float_as_uint(x);
    return (b & 0x80000000u) ? ~b : (b | 0x80000000u);
}
__device__ __forceinline__ float dec_f32(unsigned k) {
    unsigned b = (k & 0x80000000u) ? (k & 0x7FFFFFFFu) : ~k;
    return __uint_as_float(b);
}

// Init: out[n][:] = bias, per-dst max = encoded(-inf)=0, sum = 0.
__global__ void init_out_kernel(float* __restrict__ out,
                                const float* __restrict__ bias, int n) {
    int i = (int)(blockIdx.x * blockDim.x + threadIdx.x);
    if (i < n * DIM) out[i] = bias[i & (DIM - 1)];
}
__global__ void init_ms_kernel(unsigned* __restrict__ menc,
                               float* __restrict__ ssum, int n) {
    int i = (int)(blockIdx.x * blockDim.x + threadIdx.x);
    if (i < n) { menc[i] = 0u; ssum[i] = 0.f; }
}

// ---------------------------------------------------------------------------
// Pass 1: one wave per edge. logit = att . leaky_relu(xl[src] + xr[dst]);
// segment max via atomicMax on order-preserving uints.
// 32 lanes x float4 = 128 floats, fully coalesced 16B/lane.
// ---------------------------------------------------------------------------
__global__ __launch_bounds__(256) void edge_logit_kernel(
    const float* __restrict__ xl, const float* __restrict__ xr,
    const int* __restrict__ src, const int* __restrict__ dst,
    const float* __restrict__ att,
    float* __restrict__ logit, unsigned* __restrict__ menc, int nedge)
{
    const int e    = (int)((blockIdx.x * blockDim.x + threadIdx.x) >> 5);
    const int lane = (int)(threadIdx.x & 31);
    if (e >= nedge) return;
    const int s = src[e], d = dst[e];
    const float4 a  = *(const float4*)(xl + (size_t)s * DIM + lane * 4);
    const float4 b  = *(const float4*)(xr + (size_t)d * DIM + lane * 4);
    const float4 at = *(const float4*)(att + lane * 4);
    float z, sum = 0.f;
    z = a.x + b.x; z = (z > 0.f) ? z : SLOPE * z; sum += z * at.x;
    z = a.y + b.y; z = (z > 0.f) ? z : SLOPE * z; sum += z * at.y;
    z = a.z + b.z; z = (z > 0.f) ? z : SLOPE * z; sum += z * at.z;
    z = a.w + b.w; z = (z > 0.f) ? z : SLOPE * z; sum += z * at.w;
#pragma unroll
    for (int off = 16; off > 0; off >>= 1) sum += __shfl_xor(sum, off, 32);
    if (lane == 0) {
        logit[e] = sum;
        atomicMax(menc + d, enc_f32(sum));
    }
}

// Pass 2: one thread per edge. e = exp(logit - max[dst]); sum[dst] += e.
__global__ void edge_exp_kernel(
    const float* __restrict__ logit, const int* __restrict__ dst,
    const unsigned* __restrict__ menc, float* __restrict__ eexp,
    float* __restrict__ ssum, int nedge)
{
    int e = (int)(blockIdx.x * blockDim.x + threadIdx.x);
    if (e >= nedge) return;
    const int d = dst[e];
    const float v = expf(logit[e] - dec_f32(menc[d]));
    eexp[e] = v;
    atomicAdd(ssum + d, v);
}

// Pass 3: one wave per edge. out[dst] += alpha * xl[src].
__global__ __launch_bounds__(256) void edge_scatter_kernel(
    const float* __restrict__ xl,
    const int* __restrict__ src, const int* __restrict__ dst,
    const float* __restrict__ eexp, const float* __restrict__ ssum,
    float* __restrict__ out, int nedge)
{
    const int e    = (int)((blockIdx.x * blockDim.x + threadIdx.x) >> 5);
    const int lane = (int)(threadIdx.x & 31);
    if (e >= nedge) return;
    const int s = src[e], d = dst[e];
    const float alpha = eexp[e] / (ssum[d] + 1e-16f);
    const float4 a = *(const float4*)(xl + (size_t)s * DIM + lane * 4);
    float* o = out + (size_t)d * DIM + lane * 4;
    atomicAdd(o + 0, alpha * a.x);
    atomicAdd(o + 1, alpha * a.y);
    atomicAdd(o + 2, alpha * a.z);
    atomicAdd(o + 3, alpha * a.w);
}

// ---------------------------------------------------------------------------
// Host side
// ---------------------------------------------------------------------------
static inline void launch_gemm(const float* A, const float* W, const float* b,
                               float* C, int rows, int K, int ncols, int relu,
                               hipStream_t stream) {
    long waves = (long)(rows / 16);
    long threads = waves * 32;
    int blocks = (int)((threads + 127) / 128);
    if (ncols == 64)
        gemm_bias_kernel<4><<<blocks, 128, 0, stream>>>(A, W, b, C, rows, K, relu);
    else
        gemm_bias_kernel<8><<<blocks, 128, 0, stream>>>(A, W, b, C, rows, K, relu);
}

extern "C" void kernel_launch(void* const* d_in, const int* in_sizes, int n_in,
                              void* d_out, int out_size, void* d_ws, size_t ws_size,
                              hipStream_t stream) {
    (void)in_sizes; (void)n_in; (void)out_size; (void)ws_size;

    const float* x_user = (const float*)d_in[0];
    const float* x_item = (const float*)d_in[1];
    const int*   e_u2i  = (const int*)d_in[2];   // [2, E]: row0 = src, row1 = dst
    const int*   e_i2u  = (const int*)d_in[3];
    const float* W1_u = (const float*)d_in[4],  *b1_u = (const float*)d_in[5];
    const float* W2_u = (const float*)d_in[6],  *b2_u = (const float*)d_in[7];
    const float* W1_i = (const float*)d_in[8],  *b1_i = (const float*)d_in[9];
    const float* W2_i = (const float*)d_in[10], *b2_i = (const float*)d_in[11];
    const float* Wl_u2i = (const float*)d_in[12], *bl_u2i = (const float*)d_in[13];
    const float* Wr_u2i = (const float*)d_in[14], *br_u2i = (const float*)d_in[15];
    const float* att_u2i = (const float*)d_in[16], *bias_u2i = (const float*)d_in[17];
    const float* Wl_i2u = (const float*)d_in[18], *bl_i2u = (const float*)d_in[19];
    const float* Wr_i2u = (const float*)d_in[20], *br_i2u = (const float*)d_in[21];
    const float* att_i2u = (const float*)d_in[22], *bias_i2u = (const float*)d_in[23];

    float* out_user = (float*)d_out;                         // [N, 128]
    float* out_item = (float*)d_out + (size_t)N_NODES * DIM; // [N, 128]

    // workspace carve-up (floats)
    float* ws   = (float*)d_ws;
    float* hu   = ws;                              // N*128
    float* hi   = hu + (size_t)N_NODES * DIM;      // N*128
    float* tmp  = hi + (size_t)N_NODES * DIM;      // N*64
    float* xl   = tmp + (size_t)N_NODES * HID;     // N*128
    float* xr   = xl + (size_t)N_NODES * DIM;      // N*128
    float* logit = xr + (size_t)N_NODES * DIM;     // E
    float* eexp  = logit + NEDGE;                  // E
    unsigned* menc = (unsigned*)(eexp + NEDGE);    // N
    float* ssum  = (float*)(menc + N_NODES);       // N

    const int blkN    = (N_NODES + 255) / 256;
    const int blkND   = (N_NODES * DIM + 255) / 256;
    const int blkE    = (NEDGE + 255) / 256;
    const int blkEw   = (NEDGE * 32 + 255) / 256;  // wave-per-edge kernels

    // ---- node MLPs (WMMA GEMMs) ----
    launch_gemm(x_user, W1_u, b1_u, tmp, N_NODES, DIM, HID, 1, stream);
    launch_gemm(tmp,    W2_u, b2_u, hu,  N_NODES, HID, DIM, 0, stream);
    launch_gemm(x_item, W1_i, b1_i, tmp, N_NODES, DIM, HID, 1, stream);
    launch_gemm(tmp,    W2_i, b2_i, hi,  N_NODES, HID, DIM, 0, stream);

    // ---- relation u2i: users -> items, writes out_item ----
    {
        const int* src = e_u2i;           // users
        const int* dst = e_u2i + NEDGE;   // items
        launch_gemm(hu, Wl_u2i, bl_u2i, xl, N_NODES, DIM, DIM, 0, stream);
        launch_gemm(hi, Wr_u2i, br_u2i, xr, N_NODES, DIM, DIM, 0, stream);
        init_out_kernel<<<blkND, 256, 0, stream>>>(out_item, bias_u2i, N_NODES);
        init_ms_kernel<<<blkN, 256, 0, stream>>>(menc, ssum, N_NODES);
        edge_logit_kernel<<<blkEw, 256, 0, stream>>>(xl, xr, src, dst, att_u2i,
                                                     logit, menc, NEDGE);
        edge_exp_kernel<<<blkE, 256, 0, stream>>>(logit, dst, menc, eexp, ssum, NEDGE);
        edge_scatter_kernel<<<blkEw, 256, 0, stream>>>(xl, src, dst, eexp, ssum,
                                                       out_item, NEDGE);
    }

    // ---- relation i2u: items -> users, writes out_user ----
    {
        const int* src = e_i2u;           // items
        const int* dst = e_i2u + NEDGE;   // users
        launch_gemm(hi, Wl_i2u, bl_i2u, xl, N_NODES, DIM, DIM, 0, stream);
        launch_gemm(hu, Wr_i2u, br_i2u, xr, N_NODES, DIM, DIM, 0, stream);
        init_out_kernel<<<blkND, 256, 0, stream>>>(out_user, bias_i2u, N_NODES);
        init_ms_kernel<<<blkN, 256, 0, stream>>>(menc, ssum, N_NODES);
        edge_logit_kernel<<<blkEw, 256, 0, stream>>>(xl, xr, src, dst, att_i2u,
                                                     logit, menc, NEDGE);
        edge_exp_kernel<<<blkE, 256, 0, stream>>>(logit, dst, menc, eexp, ssum, NEDGE);
        edge_scatter_kernel<<<blkEw, 256, 0, stream>>>(xl, src, dst, eexp, ssum,
                                                       out_user, NEDGE);
    }
}